// RoIHeads_10161892622993
// MI455X (gfx1250) — compile-verified
//
#include <hip/hip_runtime.h>
#include <hip/hip_bf16.h>

// Problem constants (from reference setup_inputs)
constexpr int B = 8;
constexpr int P = 8000;
constexpr int G = 100;
constexpr int C = 81;
constexpr int T = P + G;          // 8100 rows per batch
constexpr int BDIM = 128;         // rows per block (4 waves)
constexpr int BLKS_X = (T + BDIM - 1) / BDIM;   // 64
constexpr int NB = BLKS_X * B;    // 512 partial slots (divisible by 64)

constexpr float FG_THRESH = 0.5f;
constexpr float BG_THRESH = 0.4f;
constexpr float SL1_BETA = 1.0f / 9.0f;

typedef __attribute__((ext_vector_type(2))) float v2f;
typedef __attribute__((ext_vector_type(8))) float v8f;

__device__ __forceinline__ float smooth_l1(float d) {
    float ad = fabsf(d);
    return (ad < SL1_BETA) ? 4.5f * ad * ad : ad - 0.5f * SL1_BETA;
}

// ---------------------------------------------------------------------------
// Stage 1: fused match + targets + per-row losses + per-block reduction.
// grid = (BLKS_X, B), block = 128. One block handles 128 rows of one batch.
// ---------------------------------------------------------------------------
__global__ __launch_bounds__(BDIM) void roiheads_stage1(
    const float* __restrict__ props,   // [B,P,4]
    const float* __restrict__ gtb,     // [B,G,4]
    const int*   __restrict__ gtl,     // [B,G]
    const float* __restrict__ logits,  // [B,T,C]
    const float* __restrict__ breg,    // [B,T,C*4]
    const float* __restrict__ bsc,     // [B,T]
    float* __restrict__ partials)      // [5, NB]
{
    __shared__ float  sL[BDIM * C];    // 41472 B logits staging (stride 81 odd -> conflict free)
    __shared__ float4 sGT[G];
    __shared__ int    sGTL[G];

    const int tid = threadIdx.x;
    const int b   = blockIdx.y;
    const int t0  = blockIdx.x * BDIM;
    const int t   = t0 + tid;

    // GT boxes + labels of this batch into LDS
    if (tid < G) {
        sGT[tid]  = ((const float4*)gtb)[b * G + tid];
        sGTL[tid] = gtl[b * G + tid];
    }

    // Coalesced staging of this block's logits tile (contiguous region)
    const int nrows = min(BDIM, T - t0);
    const long long lbase = ((long long)b * T + t0) * C;
    for (int i = tid; i < nrows * C; i += BDIM) sL[i] = logits[lbase + i];
    __syncthreads();

    float c_valid = 0.f, c_pos = 0.f, s_nll = 0.f, s_sl1 = 0.f, s_sc = 0.f;

    if (t < T) {
        // all_props = concat(proposals, gt_boxes)
        float4 pr = (t < P) ? ((const float4*)props)[b * P + t] : sGT[t - P];
        const float pw = pr.z - pr.x, ph = pr.w - pr.y;
        const float areaP = pw * ph;

        // argmax_G IoU (strict > keeps first max, matching jnp.argmax)
        float best = -1.0f; int bidx = 0;
        #pragma unroll 4
        for (int g = 0; g < G; ++g) {
            float4 gb = sGT[g];                       // broadcast, no conflicts
            float areaG = (gb.z - gb.x) * (gb.w - gb.y);
            float lx = fmaxf(gb.x, pr.x), ly = fmaxf(gb.y, pr.y);
            float rx = fminf(gb.z, pr.z), ry = fminf(gb.w, pr.w);
            float w = fmaxf(rx - lx, 0.f), h = fmaxf(ry - ly, 0.f);
            float inter = w * h;
            float iou = inter / (areaG + areaP - inter);
            if (iou > best) { best = iou; bidx = g; }
        }

        int label = (best < BG_THRESH) ? 0 : ((best < FG_THRESH) ? -1 : sGTL[bidx]);
        const bool valid = label >= 0;
        const bool pos   = label > 0;
        const int  lab   = label > 0 ? label : 0;      // safe_lab (labels <= C-1 already)

        // Prefetch the scattered class-specific regression deltas now; the
        // long log-sum-exp scan below hides the latency. (global_prefetch_b8)
        const long long roff = (((long long)b * T + t) * C + lab) * 4;
        __builtin_prefetch(breg + roff, 0, 1);

        // Cross-entropy from LDS tile
        const float* row = &sL[tid * C];
        float m = row[0];
        #pragma unroll 8
        for (int c = 1; c < C; ++c) m = fmaxf(m, row[c]);
        float se = 0.f;
        #pragma unroll 8
        for (int c = 0; c < C; ++c) se += __expf(row[c] - m);
        float nll = m + __logf(se) - row[lab];

        // Regression targets: encode(matched_gt, proposal), weights (10,10,5,5)
        float4 gb = sGT[bidx];
        float gw = gb.z - gb.x, gh = gb.w - gb.y;
        float tx = 10.f * ((gb.x + 0.5f * gw) - (pr.x + 0.5f * pw)) / pw;
        float ty = 10.f * ((gb.y + 0.5f * gh) - (pr.y + 0.5f * ph)) / ph;
        float tw = 5.f * __logf(gw / pw);
        float th = 5.f * __logf(gh / ph);

        float4 pd = *(const float4*)(breg + roff);     // 16B aligned gather
        float sl1 = smooth_l1(pd.x - tx) + smooth_l1(pd.y - ty) +
                    smooth_l1(pd.z - tw) + smooth_l1(pd.w - th);

        // score target == matched max-IoU (elementwise IoU of matched gt)
        float sdiff = fabsf(bsc[b * T + t] - best);

        c_valid = valid ? 1.f : 0.f;
        c_pos   = pos   ? 1.f : 0.f;
        s_nll   = valid ? nll  : 0.f;
        s_sl1   = pos   ? sl1  : 0.f;
        s_sc    = pos   ? sdiff: 0.f;
    }

    // Deterministic in-block tree reduction (reuse logits LDS)
    __syncthreads();
    sL[0 * BDIM + tid] = c_valid;
    sL[1 * BDIM + tid] = c_pos;
    sL[2 * BDIM + tid] = s_nll;
    sL[3 * BDIM + tid] = s_sl1;
    sL[4 * BDIM + tid] = s_sc;
    __syncthreads();
    for (int s = BDIM / 2; s > 0; s >>= 1) {
        if (tid < s) {
            #pragma unroll
            for (int q = 0; q < 5; ++q) sL[q * BDIM + tid] += sL[q * BDIM + tid + s];
        }
        __syncthreads();
    }
    if (tid == 0) {
        const int blk = blockIdx.y * gridDim.x + blockIdx.x;
        #pragma unroll
        for (int q = 0; q < 5; ++q) partials[q * NB + blk] = sL[q * BDIM];
    }
}

// ---------------------------------------------------------------------------
// Stage 2: one wave32. Deterministic final reduction via chained
// V_WMMA_F32_16X16X4_F32 with A = ones(16x4): D accumulates column sums of
// each 4x16 B tile; grand total = sum of 16 column sums (shfl within half).
// EXEC is all-ones for the whole wave (WMMA requirement).
// ---------------------------------------------------------------------------
__global__ __launch_bounds__(32) void roiheads_stage2(
    const float* __restrict__ partials, float* __restrict__ out)
{
    const int lane = threadIdx.x;
    v2f a; a.x = 1.0f; a.y = 1.0f;     // A = all-ones 16x4

    float tot[5];
    #pragma unroll
    for (int q = 0; q < 5; ++q) {
        const float* p = partials + q * NB;
        v8f acc = {};
        #pragma unroll
        for (int tI = 0; tI < NB / 64; ++tI) {   // 8 tiles of 64 values
            const int idx = tI * 64 + lane * 2;
            v2f bvec; bvec.x = p[idx]; bvec.y = p[idx + 1];
            acc = __builtin_amdgcn_wmma_f32_16x16x4_f32(
                false, a, false, bvec, (short)0, acc, false, false);
        }
        // acc[0] in lane n (and lane 16+n) = column sum S_n; fold 16 columns.
        float tq = acc[0];
        tq += __shfl_xor(tq, 1, 32);
        tq += __shfl_xor(tq, 2, 32);
        tq += __shfl_xor(tq, 4, 32);
        tq += __shfl_xor(tq, 8, 32);
        tot[q] = tq;
    }

    if (lane == 0) {
        float n_valid = fmaxf(tot[0], 1.0f);
        float n_pos   = fmaxf(tot[1], 1.0f);
        out[0] = tot[2] / n_valid;   // cls_loss
        out[1] = tot[3] / n_valid;   // box_loss (note: / n_valid per reference)
        out[2] = tot[4] / n_pos;     // score_loss
    }
}

extern "C" void kernel_launch(void* const* d_in, const int* in_sizes, int n_in,
                              void* d_out, int out_size, void* d_ws, size_t ws_size,
                              hipStream_t stream) {
    const float* props  = (const float*)d_in[0];
    const float* gtb    = (const float*)d_in[1];
    const int*   gtl    = (const int*)d_in[2];
    const float* logits = (const float*)d_in[3];
    const float* breg   = (const float*)d_in[4];
    const float* bsc    = (const float*)d_in[5];
    float* partials = (float*)d_ws;              // 5*NB floats = 10 KB

    dim3 grid(BLKS_X, B);
    roiheads_stage1<<<grid, BDIM, 0, stream>>>(props, gtb, gtl, logits, breg, bsc, partials);
    roiheads_stage2<<<1, 32, 0, stream>>>(partials, (float*)d_out);
}